// Flow_SNet_multi_71227737637094
// MI455X (gfx1250) — compile-verified
//
#include <hip/hip_runtime.h>
#include <hip/hip_bf16.h>

// ---- problem constants (from setup_inputs) ----
constexpr int BB = 2, CH = 8, DD = 64, HH = 128, WW = 128;
constexpr int HW  = HH * WW;          // 16384
constexpr int DHW = DD * HH * WW;     // 1048576

typedef __attribute__((ext_vector_type(2))) float v2f;
typedef __attribute__((ext_vector_type(8))) float v8f;

// ---------------- enc conv: 1 -> 8 ch, 3x3x3, ReLU ----------------
__global__ void enc_conv(const float* __restrict__ slices,
                         const float* __restrict__ We,
                         float* __restrict__ feats) {
  int i = blockIdx.x * blockDim.x + threadIdx.x;
  if (i >= BB * DHW) return;
  int x = i % WW, y = (i / WW) % HH, z = (i / HW) % DD, b = i / DHW;
  const float* xs = slices + (size_t)b * 2 * DHW;   // channel 0 of slices
  float in[27];
#pragma unroll
  for (int dz = 0; dz < 3; ++dz)
#pragma unroll
    for (int dy = 0; dy < 3; ++dy)
#pragma unroll
      for (int dx = 0; dx < 3; ++dx) {
        int zz = z + dz - 1, yy = y + dy - 1, xx = x + dx - 1;
        bool v = (unsigned)zz < DD && (unsigned)yy < HH && (unsigned)xx < WW;
        in[(dz * 3 + dy) * 3 + dx] = v ? xs[(size_t)zz * HW + yy * WW + xx] : 0.f;
      }
#pragma unroll
  for (int co = 0; co < 8; ++co) {
    float s = 0.f;
#pragma unroll
    for (int k = 0; k < 27; ++k) s = fmaf(We[co * 27 + k], in[k], s);
    feats[((size_t)(b * CH + co)) * DHW + (size_t)z * HW + y * WW + x] = fmaxf(s, 0.f);
  }
}

// ---------------- flow conv: 8 -> 3 ch, 3x3x3, tanh * 4 ----------------
__global__ void flow_conv(const float* __restrict__ feats,
                          const float* __restrict__ Wf,
                          float* __restrict__ flow) {
  int i = blockIdx.x * blockDim.x + threadIdx.x;
  if (i >= BB * DHW) return;
  int x = i % WW, y = (i / WW) % HH, z = (i / HW) % DD, b = i / DHW;
  float acc0 = 0.f, acc1 = 0.f, acc2 = 0.f;
  const float* fb = feats + (size_t)b * CH * DHW;
  for (int t = 0; t < 27; ++t) {
    int dz = t / 9, r = t - dz * 9, dy = r / 3, dx = r - dy * 3;
    int zz = z + dz - 1, yy = y + dy - 1, xx = x + dx - 1;
    if ((unsigned)zz >= DD || (unsigned)yy >= HH || (unsigned)xx >= WW) continue;
    size_t sp = (size_t)zz * HW + yy * WW + xx;
#pragma unroll
    for (int ci = 0; ci < 8; ++ci) {
      float f = fb[(size_t)ci * DHW + sp];
      acc0 = fmaf(Wf[(0 * 8 + ci) * 27 + t], f, acc0);
      acc1 = fmaf(Wf[(1 * 8 + ci) * 27 + t], f, acc1);
      acc2 = fmaf(Wf[(2 * 8 + ci) * 27 + t], f, acc2);
    }
  }
  size_t o = (size_t)b * 3 * DHW + (size_t)z * HW + y * WW + x;
  flow[o]            = tanhf(acc0) * 4.0f;
  flow[o + DHW]      = tanhf(acc1) * 4.0f;
  flow[o + 2 * DHW]  = tanhf(acc2) * 4.0f;
}

// ---------------- zero splat buffer + denom ----------------
__global__ void zero_ws(float* __restrict__ splat, unsigned* __restrict__ denomBits, size_t n) {
  size_t i = (size_t)blockIdx.x * blockDim.x + threadIdx.x;
  size_t stride = (size_t)gridDim.x * blockDim.x;
  for (; i < n; i += stride) splat[i] = 0.f;
  if (blockIdx.x == 0 && threadIdx.x < BB) denomBits[threadIdx.x] = 0u;
}

// ---------------- splat: 12 unique trilinear corners (PSF folded), 9 ch ----------------
__global__ void splat_k(const float* __restrict__ feats,
                        const float* __restrict__ slices,
                        const float* __restrict__ flow,
                        float* __restrict__ splat) {
  int i = blockIdx.x * blockDim.x + threadIdx.x;
  if (i >= BB * DHW) return;
  int x = i % WW, y = (i / WW) % HH, z = (i / HW) % DD, b = i / DHW;
  size_t sp = (size_t)z * HW + y * WW + x;

  float m = slices[((size_t)b * 2 + 1) * DHW + sp];
  float v[9];
#pragma unroll
  for (int c = 0; c < 8; ++c) v[c] = feats[((size_t)(b * CH + c)) * DHW + sp] * m;
  v[8] = m;

  size_t fo = (size_t)b * 3 * DHW + sp;
  float pz = (float)z + flow[fo];
  float py = (float)y + flow[fo + DHW];
  float px = (float)x + flow[fo + 2 * DHW];

  float z0f = floorf(pz), y0f = floorf(py), x0f = floorf(px);
  float fz = pz - z0f, fy = py - y0f, fx = px - x0f;
  int z0 = (int)z0f, y0 = (int)y0f, x0 = (int)x0f;

  // PSF taps at z-offsets {0,+1}, weight 0.5 each -> 3 z-levels:
  float wz[3] = {0.5f * (1.f - fz), 0.5f, 0.5f * fz};
  float wy[2] = {1.f - fy, fy};
  float wx[2] = {1.f - fx, fx};

  float* sb = splat + (size_t)b * 9 * DHW;
#pragma unroll
  for (int iz = 0; iz < 3; ++iz) {
    int zi = z0 + iz;
    if ((unsigned)zi >= DD) continue;
#pragma unroll
    for (int iy = 0; iy < 2; ++iy) {
      int yi = y0 + iy;
      if ((unsigned)yi >= HH) continue;
#pragma unroll
      for (int ix = 0; ix < 2; ++ix) {
        int xi = x0 + ix;
        if ((unsigned)xi >= WW) continue;
        float w = wz[iz] * wy[iy] * wx[ix];
        size_t base = (size_t)zi * HW + yi * WW + xi;
#pragma unroll
        for (int c = 0; c < 9; ++c)
          unsafeAtomicAdd(&sb[(size_t)c * DHW + base], v[c] * w);
      }
    }
  }
}

// ---------------- denom = max over splat[:, 8] (non-negative -> uint-max trick) ----------------
__global__ void denom_max(const float* __restrict__ splat, unsigned* __restrict__ denomBits) {
  int b = blockIdx.y;
  const float* d = splat + ((size_t)b * 9 + 8) * DHW;
  float mx = 0.f;
  for (size_t i = (size_t)blockIdx.x * blockDim.x + threadIdx.x; i < (size_t)DHW;
       i += (size_t)gridDim.x * blockDim.x)
    mx = fmaxf(mx, d[i]);
  __shared__ float red[256];
  red[threadIdx.x] = mx;
  __syncthreads();
  for (int s = 128; s > 0; s >>= 1) {
    if (threadIdx.x < s) red[threadIdx.x] = fmaxf(red[threadIdx.x], red[threadIdx.x + s]);
    __syncthreads();
  }
  if (threadIdx.x == 0) atomicMax(&denomBits[b], __float_as_uint(red[0]));
}

// ---------------- dec conv 8->8 via V_WMMA_F32_16X16X4_F32 (LDS-resident implicit GEMM) ----
// Block = 4 waves = one (b,z,y) row, 64-wide x strip (wave w -> 16 x-positions).
// Input tile (3z x 3y x 66x x 8ci) staged to LDS with GLOBAL_LOAD_ASYNC_TO_LDS_B32
// (clamped addresses; OOB entries patched to zero so boundary handling costs nothing
// in the fully-unrolled 54-step WMMA loop and EXEC stays all-ones for every WMMA).
// 1/denom folded into epilogue; ReLU at store.
constexpr int TZY = 9;          // 3*3 zy-planes
constexpr int TX  = 66;         // 64 + 2 halo
constexpr int TCP = 10;         // 8 ci padded to 10 (keeps b64 pairs 8B-aligned)
constexpr int NSTAGE = TZY * TX * 8;   // elements actually staged

__global__ __launch_bounds__(128) void dec_wmma(const float* __restrict__ splat,
                                                const float* __restrict__ Wd,
                                                const unsigned* __restrict__ denomBits,
                                                float* __restrict__ dec) {
  __shared__ float wlds[54 * 64];          // B-matrix per k-step (13.8 KB)
  __shared__ float tile[TZY * TX * TCP];   // A input tile      (23.8 KB)
  int tid = threadIdx.x;

  // B layout (4x16 f32, 2 VGPR/lane): lane l holds (K=k0,N=l%16),(K=k0+1,N), k0=(l/16)*2
  for (int i = tid; i < 54 * 64; i += 128) {
    int s = i >> 6, l2 = i & 63, l = l2 >> 1, j = l2 & 1;
    int t = s >> 1, cig = (s & 1) * 4;
    int ci = cig + ((l >> 4) << 1) + j, co = l & 15;
    wlds[i] = (co < 8) ? Wd[(co * 8 + ci) * 27 + t] : 0.f;
  }

  int g = blockIdx.x;                       // B*D*H*(W/64) = 32768 blocks
  int xs = g & 1, y = (g >> 1) & 127, z = (g >> 8) & 63, b = g >> 14;
  int xbase = xs << 6;
  const float* sp = splat + (size_t)b * 9 * DHW;

  // ---- async stage of the input tile (clamped addresses, every lane valid) ----
  for (int idx = tid; idx < NSTAGE; idx += 128) {
    int zy  = idx / (TX * 8);
    int rem = idx - zy * (TX * 8);
    int xl = rem >> 3, ci = rem & 7;
    int zz = z + zy / 3 - 1, yy = y + zy % 3 - 1, xx = xbase + xl - 1;
    int zc = min(max(zz, 0), DD - 1);
    int yc = min(max(yy, 0), HH - 1);
    int xc = min(max(xx, 0), WW - 1);
    const float* gp = sp + (size_t)ci * DHW + (size_t)zc * HW + yc * WW + xc;
    unsigned la = (unsigned)(uintptr_t)&tile[(zy * TX + xl) * TCP + ci];
    asm volatile("global_load_async_to_lds_b32 %0, %1, off"
                 :: "v"(la), "v"((unsigned long long)(uintptr_t)gp)
                 : "memory");
  }
  asm volatile("s_wait_asynccnt 0x0" ::: "memory");
  // patch out-of-bounds entries to zero (this thread's own staged entries)
  for (int idx = tid; idx < NSTAGE; idx += 128) {
    int zy  = idx / (TX * 8);
    int rem = idx - zy * (TX * 8);
    int xl = rem >> 3, ci = rem & 7;
    int zz = z + zy / 3 - 1, yy = y + zy % 3 - 1, xx = xbase + xl - 1;
    if ((unsigned)zz >= (unsigned)DD || (unsigned)yy >= (unsigned)HH ||
        (unsigned)xx >= (unsigned)WW)
      tile[(zy * TX + xl) * TCP + ci] = 0.f;
  }
  __syncthreads();

  // ---- fully-unrolled 54-step WMMA GEMM: all LDS offsets are immediates ----
  int lane = tid & 31;
  int wv   = tid >> 5;
  int m = lane & 15, half = lane >> 4;
  int xloc = (wv << 4) + m;                 // 0..63 within strip
  v8f acc = {};

#pragma unroll
  for (int dz = 0; dz < 3; ++dz)
#pragma unroll
    for (int dy = 0; dy < 3; ++dy)
#pragma unroll
      for (int dx = 0; dx < 3; ++dx) {
        int zy = dz * 3 + dy;
        int t  = zy * 3 + dx;
#pragma unroll
        for (int cg = 0; cg < 2; ++cg) {
          int ci = cg * 4 + half * 2;       // A layout: lane half selects K pair
          v2f a  = *(const v2f*)&tile[(zy * TX + xloc + dx) * TCP + ci];
          v2f bm = *(const v2f*)&wlds[(t * 2 + cg) * 64 + lane * 2];
          acc = __builtin_amdgcn_wmma_f32_16x16x4_f32(false, a, false, bm,
                                                      (short)0, acc, false, false);
        }
      }

  float den  = __uint_as_float(denomBits[b]);
  float rden = 1.0f / fmaxf(den, 1e-6f);

  int co = m;                               // D layout: N = lane%16
  if (co < 8) {
    float* op = dec + ((size_t)(b * 8 + co)) * DHW + (size_t)z * HW + (size_t)y * WW
              + xbase + (wv << 4) + half * 8;   // VGPR r -> M = r + half*8
#pragma unroll
    for (int r = 0; r < 8; ++r) op[r] = fmaxf(acc[r] * rden, 0.f);
  }
}

// ---------------- trilinear gather warp ----------------
__global__ void sample_k(const float* __restrict__ dec,
                         const float* __restrict__ flow,
                         float* __restrict__ out) {
  int i = blockIdx.x * blockDim.x + threadIdx.x;
  if (i >= BB * DHW) return;
  int x = i % WW, y = (i / WW) % HH, z = (i / HW) % DD, b = i / DHW;
  size_t sp = (size_t)z * HW + y * WW + x;

  size_t fo = (size_t)b * 3 * DHW + sp;
  float pz = (float)z + flow[fo];
  float py = (float)y + flow[fo + DHW];
  float px = (float)x + flow[fo + 2 * DHW];

  float z0f = floorf(pz), y0f = floorf(py), x0f = floorf(px);
  float fz = pz - z0f, fy = py - y0f, fx = px - x0f;
  int z0 = (int)z0f, y0 = (int)y0f, x0 = (int)x0f;

  float o[8];
#pragma unroll
  for (int c = 0; c < 8; ++c) o[c] = 0.f;

  const float* db = dec + (size_t)b * 8 * DHW;
#pragma unroll
  for (int dz = 0; dz < 2; ++dz) {
    int zi = z0 + dz;
    if ((unsigned)zi >= DD) continue;
    float wz = dz ? fz : 1.f - fz;
#pragma unroll
    for (int dy = 0; dy < 2; ++dy) {
      int yi = y0 + dy;
      if ((unsigned)yi >= HH) continue;
      float wy = dy ? fy : 1.f - fy;
#pragma unroll
      for (int dx = 0; dx < 2; ++dx) {
        int xi = x0 + dx;
        if ((unsigned)xi >= WW) continue;
        float w = wz * wy * (dx ? fx : 1.f - fx);
        size_t base = (size_t)zi * HW + yi * WW + xi;
#pragma unroll
        for (int c = 0; c < 8; ++c) o[c] = fmaf(db[(size_t)c * DHW + base], w, o[c]);
      }
    }
  }
#pragma unroll
  for (int c = 0; c < 8; ++c)
    out[((size_t)(b * 8 + c)) * DHW + sp] = o[c];
}

extern "C" void kernel_launch(void* const* d_in, const int* in_sizes, int n_in,
                              void* d_out, int out_size, void* d_ws, size_t ws_size,
                              hipStream_t stream) {
  const float* slices = (const float*)d_in[0];   // (2,2,64,128,128)
  const float* We     = (const float*)d_in[1];   // (8,1,3,3,3)
  const float* Wf     = (const float*)d_in[2];   // (3,8,3,3,3)
  const float* Wd     = (const float*)d_in[3];   // (8,8,3,3,3)
  float* out = (float*)d_out;

  float* ws    = (float*)d_ws;
  float* feats = ws;                                    // B*8*DHW  (reused for dec)
  float* flow  = feats + (size_t)BB * CH * DHW;         // B*3*DHW
  float* splat = flow  + (size_t)BB * 3 * DHW;          // B*9*DHW
  unsigned* denomBits = (unsigned*)(splat + (size_t)BB * 9 * DHW);
  float* dec = feats;                                   // feats dead after splat

  const int nvox = BB * DHW;
  const int blk = 256, grd = (nvox + blk - 1) / blk;

  enc_conv <<<grd, blk, 0, stream>>>(slices, We, feats);
  flow_conv<<<grd, blk, 0, stream>>>(feats, Wf, flow);
  zero_ws  <<<4096, blk, 0, stream>>>(splat, denomBits, (size_t)BB * 9 * DHW);
  splat_k  <<<grd, blk, 0, stream>>>(feats, slices, flow, splat);
  denom_max<<<dim3(512, BB), blk, 0, stream>>>(splat, denomBits);
  // strips = B*D*H*(W/64) = 32768 blocks, 4 waves/block
  dec_wmma <<<32768, 128, 0, stream>>>(splat, Wd, denomBits, dec);
  sample_k <<<grd, blk, 0, stream>>>(dec, flow, out);
}